// SelectiveSSM_8100308320555
// MI455X (gfx1250) — compile-verified
//
#include <hip/hip_runtime.h>
#include <hip/hip_bf16.h>

// ---------------------------------------------------------------------------
// Mamba selective-SSM forward for MI455X (gfx1250), fp32 end-to-end.
// Pipeline: gemm_in -> conv+silu -> gemm_xproj -> gemm_dt(+softplus) ->
//           scan(fused D-skip + silu gate) -> gemm_out
// GEMMs: V_WMMA_F32_16X16X4_F32, 2x2 register blocking per wave (32x32
// macro-tile), compile-time N/K/LDA (immediate-offset loads), and chunk-level
// double buffering so load latency is covered by the previous chunk's WMMAs.
// fp32 WMMA keeps exact reference numerics; the workload (~55 GFLOP, L2
// resident) is bandwidth/latency bound, so bf16 WMMA would buy nothing.
// ---------------------------------------------------------------------------

typedef float v2f __attribute__((ext_vector_type(2)));
typedef float v8f __attribute__((ext_vector_type(8)));

#define D_MODEL 1024
#define D_INNER 2048
#define D_STATE 16
#define DT_RANK 64
#define XPROJ_N 96   // DT_RANK + 2*D_STATE
#define SEQ_L   2048
#define NBATCH  2

// ---------------------------------------------------------------------------
// fp32 WMMA GEMM: C[M,N] = A[M,K(LDA)] * B[K,N]  (row major, dims static)
// EPI==0: plain store.  EPI==1: C = softplus(acc + bias[n])  (for delta).
// Block: 256 threads = 8 waves; each wave computes a 32x32 macro-tile
// (2x2 WMMA tiles); block tile = 256(M) x 32(N). Grid: (N/32, M/256).
// K must be a multiple of 32 (holds: 1024, 2048, 64).
// ---------------------------------------------------------------------------
template <int EPI, int N, int K, int LDA>
__global__ __launch_bounds__(256)
void wmma_gemm_f32(const float* __restrict__ A, const float* __restrict__ Bm,
                   const float* __restrict__ bias, float* __restrict__ C)
{
    const int lane = threadIdx.x & 31;
    const int wave = threadIdx.x >> 5;
    const int n0   = blockIdx.x * 32;
    const int m0   = blockIdx.y * 256 + wave * 32;

    const int ncol  = n0 + (lane & 15);
    const int kbase = (lane >> 4) << 1;          // 0 for lanes 0-15, 2 for 16-31

    // A fragments for the two M sub-tiles (a.x = K+0, a.y = K+1 per lane)
    const float* Aptr0 = A + (size_t)(m0 + (lane & 15)) * LDA + kbase;
    const float* Aptr1 = Aptr0 + (size_t)16 * LDA;
    // B fragment base (b.x = row K, b.y = row K+1)
    const float* Bptr = Bm + (size_t)kbase * N + ncol;

    v8f acc00 = {}, acc01 = {}, acc10 = {}, acc11 = {};

    v2f ca0[4], ca1[4], cb0[4], cb1[4];   // current 16-K chunk fragments
    v2f pa0[4], pa1[4], pb0[4], pb1[4];   // prefetched next chunk

    // load one 16-K chunk (4 WMMA substeps) of fragments
    auto loadc = [&](int k, v2f* A0, v2f* A1, v2f* B0, v2f* B1) {
#pragma unroll
        for (int s = 0; s < 4; ++s) {
            const int kk = 4 * s;
            A0[s] = *(const v2f*)(Aptr0 + k + kk);
            A1[s] = *(const v2f*)(Aptr1 + k + kk);
            const float* Br = Bptr + (size_t)(k + kk) * N;
            B0[s].x = Br[0];   B0[s].y = Br[N];
            B1[s].x = Br[16];  B1[s].y = Br[N + 16];
        }
    };
    // 16 WMMAs (4 substeps x 2x2 tiles) on one chunk
    auto mma = [&](const v2f* A0, const v2f* A1, const v2f* B0, const v2f* B1) {
#pragma unroll
        for (int s = 0; s < 4; ++s) {
            acc00 = __builtin_amdgcn_wmma_f32_16x16x4_f32(false, A0[s], false, B0[s],
                        (short)0, acc00, false, false);
            acc01 = __builtin_amdgcn_wmma_f32_16x16x4_f32(false, A0[s], false, B1[s],
                        (short)0, acc01, false, false);
            acc10 = __builtin_amdgcn_wmma_f32_16x16x4_f32(false, A1[s], false, B0[s],
                        (short)0, acc10, false, false);
            acc11 = __builtin_amdgcn_wmma_f32_16x16x4_f32(false, A1[s], false, B1[s],
                        (short)0, acc11, false, false);
        }
    };

    loadc(0, ca0, ca1, cb0, cb1);
#pragma unroll 1
    for (int k = 0; k < K; k += 32) {
        loadc(k + 16, pa0, pa1, pb0, pb1);       // issue next chunk's loads
        mma(ca0, ca1, cb0, cb1);                 // ...while crunching current
        if (k + 32 < K)
            loadc(k + 32, ca0, ca1, cb0, cb1);
        mma(pa0, pa1, pb0, pb1);
    }

    float bv0 = 0.f, bv1 = 0.f;
    if (EPI == 1) { bv0 = bias[ncol]; bv1 = bias[ncol + 16]; }

    // C/D layout: VGPR j -> row tile_m0 + j + 8*(lane>>4), col = ncol (+16)
    float* Crow0 = C + (size_t)(m0 + 8 * (lane >> 4)) * N + ncol;
    float* Crow1 = Crow0 + (size_t)16 * N;
#pragma unroll
    for (int j = 0; j < 8; ++j) {
        float v00 = acc00[j], v01 = acc01[j], v10 = acc10[j], v11 = acc11[j];
        if (EPI == 1) {                       // softplus(acc + b_dt)
            v00 += bv0; v01 += bv1; v10 += bv0; v11 += bv1;
            v00 = (v00 > 20.f) ? v00 : __logf(1.f + __expf(v00));
            v01 = (v01 > 20.f) ? v01 : __logf(1.f + __expf(v01));
            v10 = (v10 > 20.f) ? v10 : __logf(1.f + __expf(v10));
            v11 = (v11 > 20.f) ? v11 : __logf(1.f + __expf(v11));
        }
        Crow0[(size_t)j * N]      = v00;
        Crow0[(size_t)j * N + 16] = v01;
        Crow1[(size_t)j * N]      = v10;
        Crow1[(size_t)j * N + 16] = v11;
    }
}

// ---------------------------------------------------------------------------
// Depthwise causal conv (K=4, left pad 3) + bias + SiLU.
// xi lives in xz[..., :D_INNER] (row stride 2*D_INNER). One thread per
// (b, t, d); output xs is (B, L, D_INNER).
// ---------------------------------------------------------------------------
__global__ __launch_bounds__(256)
void conv_silu_kernel(const float* __restrict__ xz, const float* __restrict__ w,
                      const float* __restrict__ cb, float* __restrict__ xs, int L)
{
    int g  = blockIdx.x * blockDim.x + threadIdx.x;       // over B*L*D_INNER
    int d  = g % D_INNER;
    int bt = g / D_INNER;                                  // b*L + t
    int t  = bt % L;

    const float4 wv = *(const float4*)(w + d * 4);
    float wk[4] = {wv.x, wv.y, wv.z, wv.w};
    float acc = cb[d];
#pragma unroll
    for (int k = 0; k < 4; ++k) {
        int ts = t + k - 3;
        if (ts >= 0)
            acc += wk[k] * xz[((size_t)(bt + k - 3)) * (2 * D_INNER) + d];
    }
    xs[(size_t)g] = acc / (1.f + __expf(-acc));            // SiLU
}

// ---------------------------------------------------------------------------
// Selective scan: one thread per (b, d) channel; 16 states + 16 A values in
// registers; B_t/C_t staged through LDS in chunks of 8 timesteps. Fuses the
// D*x skip connection and the silu(z) gate; output may alias `delta` (each
// thread reads delta[row,d] before writing yg[row,d]).
// ---------------------------------------------------------------------------
__global__ __launch_bounds__(256)
void scan_kernel(const float* __restrict__ delta, const float* __restrict__ xp,
                 const float* __restrict__ xs, const float* __restrict__ xz,
                 const float* __restrict__ A_log, const float* __restrict__ Dp,
                 float* __restrict__ yg, int L)
{
    __shared__ float sBC[256];                 // 8 timesteps x 32 (B_t ++ C_t)

    int g = blockIdx.x * 256 + threadIdx.x;    // over B*D_INNER (b uniform/block)
    int d = g % D_INNER;
    int b = g / D_INNER;

    float Arow[D_STATE], h[D_STATE];
#pragma unroll
    for (int n = 0; n < D_STATE; ++n) {
        Arow[n] = -__expf(A_log[d * D_STATE + n]);
        h[n] = 0.f;
    }
    const float Dd = Dp[d];
    const size_t baseRow = (size_t)b * L;

    for (int t0 = 0; t0 < L; t0 += 8) {
        __syncthreads();
        {
            int tt = threadIdx.x >> 5, j = threadIdx.x & 31;
            sBC[threadIdx.x] = xp[(baseRow + t0 + tt) * XPROJ_N + DT_RANK + j];
        }
        __syncthreads();
#pragma unroll 1
        for (int tt = 0; tt < 8; ++tt) {
            size_t row = baseRow + t0 + tt;
            float dlt = delta[row * D_INNER + d];
            float xv  = xs[row * D_INNER + d];
            float zv  = xz[row * (2 * D_INNER) + D_INNER + d];
            const float* Bt = &sBC[tt * 32];
            float y = 0.f;
#pragma unroll
            for (int n = 0; n < D_STATE; ++n) {
                float dA = __expf(dlt * Arow[n]);
                h[n] = dA * h[n] + (dlt * Bt[n]) * xv;
                y += h[n] * Bt[D_STATE + n];
            }
            y += Dd * xv;                                  // skip: + D*x
            y *= zv / (1.f + __expf(-zv));                 // gate: * silu(z)
            yg[row * D_INNER + d] = y;
        }
    }
}

// ---------------------------------------------------------------------------
extern "C" void kernel_launch(void* const* d_in, const int* in_sizes, int n_in,
                              void* d_out, int out_size, void* d_ws, size_t ws_size,
                              hipStream_t stream)
{
    (void)in_sizes; (void)n_in; (void)out_size; (void)ws_size;

    const float* x       = (const float*)d_in[0];
    const float* W_in    = (const float*)d_in[1];
    const float* conv_w  = (const float*)d_in[2];
    const float* conv_b  = (const float*)d_in[3];
    const float* W_xproj = (const float*)d_in[4];
    const float* W_dt    = (const float*)d_in[5];
    const float* b_dt    = (const float*)d_in[6];
    const float* A_log   = (const float*)d_in[7];
    const float* Dp      = (const float*)d_in[8];
    const float* W_out   = (const float*)d_in[9];
    float* out = (float*)d_out;

    const int Mrows = NBATCH * SEQ_L;                     // 4096

    float* ws = (float*)d_ws;
    float* xz = ws;                                       // (B,L,2*Di)  64 MB
    float* xs = xz + (size_t)Mrows * 2 * D_INNER;         // (B,L,Di)    32 MB
    float* xp = xs + (size_t)Mrows * D_INNER;             // (B,L,96)   1.5 MB
    float* dl = xp + (size_t)Mrows * XPROJ_N;             // delta, reused as yg

    dim3 blk(256);

    // 1) xz = x @ W_in                      M=4096 N=4096 K=1024
    wmma_gemm_f32<0, 2 * D_INNER, D_MODEL, D_MODEL>
        <<<dim3(2 * D_INNER / 32, Mrows / 256), blk, 0, stream>>>(
            x, W_in, nullptr, xz);

    // 2) xs = silu(depthwise_conv(xi) + b)
    conv_silu_kernel<<<dim3((Mrows * D_INNER) / 256), blk, 0, stream>>>(
        xz, conv_w, conv_b, xs, SEQ_L);

    // 3) xp = xs @ W_xproj                  M=4096 N=96 K=2048
    wmma_gemm_f32<0, XPROJ_N, D_INNER, D_INNER>
        <<<dim3(XPROJ_N / 32, Mrows / 256), blk, 0, stream>>>(
            xs, W_xproj, nullptr, xp);

    // 4) delta = softplus(xp[:,:64] @ W_dt + b_dt)   M=4096 N=2048 K=64 lda=96
    wmma_gemm_f32<1, D_INNER, DT_RANK, XPROJ_N>
        <<<dim3(D_INNER / 32, Mrows / 256), blk, 0, stream>>>(
            xp, W_dt, b_dt, dl);

    // 5) selective scan + D-skip + silu(z) gate (writes over delta buffer)
    scan_kernel<<<dim3((NBATCH * D_INNER) / 256), blk, 0, stream>>>(
        dl, xp, xs, xz, A_log, Dp, dl, SEQ_L);

    // 6) out = yg @ W_out                   M=4096 N=1024 K=2048
    wmma_gemm_f32<0, D_MODEL, D_INNER, D_INNER>
        <<<dim3(D_MODEL / 32, Mrows / 256), blk, 0, stream>>>(
            dl, W_out, nullptr, out);
}